// First_level_net_19791209300221
// MI455X (gfx1250) — compile-verified
//
#include <hip/hip_runtime.h>

// ---------------------------------------------------------------------------
// Fused: 3x3 conv(1->4) + relu  ->  12 shifted-product channels  ->  avg-pool
// Single pass over HBM (bandwidth-bound: ~1.07 GB traffic, 23.3 TB/s floor).
// Conv is done per-tile with V_WMMA_F32_16X16X4_F32 (16 pixels x 4 ch, K=9
// taps as 3 WMMAs of K=4), feats staged in LDS with a *linear* pixel-index
// layout so the WMMA D-scatter is 8 unconditional ds_store_b32 (imm offsets),
// outputs streamed with NT stores (1 GB output cannot live in 192 MB L2).
// ---------------------------------------------------------------------------

typedef __attribute__((ext_vector_type(2))) float v2f;
typedef __attribute__((ext_vector_type(8))) float v8f;

#define HIMG 4096
#define HO   4094          // feats / mult_x spatial size
#define WP   2047          // pooled spatial size
#define TILE 32            // output tile per workgroup
#define FRD  34            // feats halo region (TILE + 2)
#define XD   36            // x halo region (FRD + 2)
#define XS   37            // x LDS row stride (pad)
#define NGRP 73            // ceil(34*34 / 16) pixel groups for WMMA
#define FCH  (NGRP * 16)   // 1168 slots per feats channel (12 pad slots)

__global__ __launch_bounds__(256)
void fused_conv_products_pool(const float* __restrict__ x,
                              const float* __restrict__ w,
                              float* __restrict__ out) {
    __shared__ float xtile[XD * XS];            // 5328 B
    __shared__ float ftile[4 * FCH];            // 18688 B, ftile[ch*1168 + fr*34 + fc]

    const int gy0 = blockIdx.y * TILE;
    const int gx0 = blockIdx.x * TILE;
    const int fbase = gy0 - 1;                  // global row of feats slot 0
    const int cbase = gx0 - 1;
    const int tid = threadIdx.x;

    // ---- Phase 1: load 36x36 x halo into LDS (OOB -> 0, never used) ----
    for (int idx = tid; idx < XD * XD; idx += 256) {
        int r = idx / XD, c = idx - r * XD;
        int gr = fbase + r, gc = cbase + c;
        float v = 0.f;
        if (gr >= 0 && gr < HIMG && gc >= 0 && gc < HIMG)
            v = x[(long)gr * HIMG + gc];
        xtile[r * XS + c] = v;
    }
    __syncthreads();

    // ---- Phase 2: WMMA conv + relu -> feats (4 ch x 1168 slots) in LDS ----
    {
        const int lane = tid & 31;
        const int wave = tid >> 5;
        const int m    = lane & 15;   // A: pixel row M ; B: column N (channel)
        const int kh   = lane >> 4;   // K half select (A/B VGPR r -> K = r + 2*kh)

        // B = weights, invariant across groups. B[K][N] = w[N][tap=K].
        v2f B[3];
#pragma unroll
        for (int t = 0; t < 3; ++t) {
            int k0 = t * 4 + 2 * kh;
            B[t].x = (k0     < 9 && m < 4) ? w[m * 9 + k0]     : 0.f;
            B[t].y = (k0 + 1 < 9 && m < 4) ? w[m * 9 + k0 + 1] : 0.f;
        }

        for (int g = wave; g < NGRP; g += 8) {
            int p = g * 16 + m;
            if (p > FRD * FRD - 1) p = FRD * FRD - 1;   // pad slots: dup pixel
            int fr = p / FRD, fc = p - fr * FRD;
            int gr = fbase + fr;  gr = gr < 0 ? 0 : (gr > HO - 1 ? HO - 1 : gr);
            int gc = cbase + fc;  gc = gc < 0 ? 0 : (gc > HO - 1 ? HO - 1 : gc);
            int lr = gr - fbase, lc = gc - cbase;       // x-tile local

            v8f acc = {};
#pragma unroll
            for (int t = 0; t < 3; ++t) {
                int k0 = t * 4 + 2 * kh;                // tap for VGPR 0
                int k1 = k0 + 1;                        // tap for VGPR 1
                v2f A;
                A.x = (k0 < 9) ? xtile[(lr + k0 / 3) * XS + lc + (k0 - (k0 / 3) * 3)] : 0.f;
                A.y = (k1 < 9) ? xtile[(lr + k1 / 3) * XS + lc + (k1 - (k1 / 3) * 3)] : 0.f;
                acc = __builtin_amdgcn_wmma_f32_16x16x4_f32(
                        false, A, false, B[t], (short)0, acc, false, false);
            }

            // D: VGPR v, lane -> pixel M = v + 8*kh, channel N = m.
            // Linear layout: no bounds check, no division -> 8 ds_store_b32.
            if (m < 4) {
                int baseoff = m * FCH + g * 16 + 8 * kh;
#pragma unroll
                for (int v = 0; v < 8; ++v)
                    ftile[baseoff + v] = fmaxf(acc[v], 0.f);
            }
        }
    }
    __syncthreads();

    // ---- Phase 3: 12 shifted products + fused 2x2 avg-pool ----
    const int qr = tid >> 4, qc = tid & 15;         // 2x2 quad per thread
    const int gi0 = gy0 + 2 * qr, gj0 = gx0 + 2 * qc;
    if (gi0 >= HO || gj0 >= HO) return;             // quads fully in or out

    float sums[12];
#pragma unroll
    for (int ch = 0; ch < 12; ++ch) sums[ch] = 0.f;

    const long POOLED = (long)12 * HO * HO;

#pragma unroll
    for (int dy = 0; dy < 2; ++dy) {
        float vals[2][12];
#pragma unroll
        for (int dx = 0; dx < 2; ++dx) {
            int i = gi0 + dy, j = gj0 + dx;
            int fr = i - fbase, fc = j - cbase;      // local, in [1,32]
            bool zi  = (i == HO - 2);                // row_ins zero row
            bool zj  = (j == HO - 2);                // col_ins zero col
            bool zsr = (j == 0);                     // col_shift_right zero
            bool zsd = (i == 0);                     // row_shift_down zero
            int ri = (i == HO - 1) ? fr : fr + 1;    // row_ins target
            int cj = (j == HO - 1) ? fc : fc + 1;    // col_ins target

            float base[4], av[4], bv[4], cv[4];
#pragma unroll
            for (int ch = 0; ch < 4; ++ch) {
                const float* F = &ftile[ch * FCH];
                base[ch] = F[fr * FRD + fc];
                if (ch < 2) {   // vertical_diagonal pattern (feats 0,1)
                    float a = F[ri * FRD + cj];
                    float b = F[ri * FRD + fc - 1];
                    float c = F[ri * FRD + fc];
                    av[ch] = (zi || zj)  ? 0.f : a;  // col_ins(row_ins)
                    bv[ch] = (zi || zsr) ? 0.f : b;  // cshr(row_ins)
                    cv[ch] = zi          ? 0.f : c;  // row_ins
                } else {        // horizontal pattern (feats 2,3)
                    float a = F[ri * FRD + cj];
                    float b = F[(fr - 1) * FRD + cj];
                    float c = F[fr * FRD + cj];
                    av[ch] = (zi || zj)  ? 0.f : a;  // row_ins(col_ins)
                    bv[ch] = (zsd || zj) ? 0.f : b;  // rshd(col_ins)
                    cv[ch] = zj          ? 0.f : c;  // col_ins
                }
            }
            float* o = vals[dx];
            o[0]  = cv[2] * base[2];   // horiz
            o[1]  = av[0] * base[0];   // fd
            o[2]  = cv[0] * base[0];   // vert
            o[3]  = bv[0] * base[0];   // sd
            o[4]  = av[2] * base[2];   // fdd
            o[5]  = bv[2] * base[2];   // sdd
            o[6]  = cv[3] * base[3];   // ohoriz
            o[7]  = av[1] * base[1];   // ofd
            o[8]  = cv[1] * base[1];   // overt
            o[9]  = bv[1] * base[1];   // osd
            o[10] = av[3] * base[3];   // ofdd
            o[11] = bv[3] * base[3];   // osdd
        }
        long rowoff = (long)(gi0 + dy) * HO + gj0;   // even -> 8B aligned
#pragma unroll
        for (int ch = 0; ch < 12; ++ch) {
            v2f pr; pr.x = vals[0][ch]; pr.y = vals[1][ch];
            __builtin_nontemporal_store(pr, (v2f*)(out + (long)ch * HO * HO + rowoff));
            sums[ch] += vals[0][ch] + vals[1][ch];
        }
    }

    {
        int pi = gi0 >> 1, pj = gj0 >> 1;
        long poff = POOLED + (long)pi * WP + pj;
#pragma unroll
        for (int ch = 0; ch < 12; ++ch)
            __builtin_nontemporal_store(sums[ch] * 0.25f,
                                        out + poff + (long)ch * WP * WP);
    }
}

extern "C" void kernel_launch(void* const* d_in, const int* in_sizes, int n_in,
                              void* d_out, int out_size, void* d_ws, size_t ws_size,
                              hipStream_t stream) {
    (void)in_sizes; (void)n_in; (void)out_size; (void)d_ws; (void)ws_size;
    const float* x = (const float*)d_in[0];   // (1,1,4096,4096) f32
    const float* w = (const float*)d_in[1];   // (4,1,3,3) f32
    float* out = (float*)d_out;               // mult_x flat ++ pooled flat
    dim3 grid((HO + TILE - 1) / TILE, (HO + TILE - 1) / TILE);  // 128 x 128
    fused_conv_products_pool<<<grid, 256, 0, stream>>>(x, w, out);
}